// FilteredLReLU_20727512170534
// MI455X (gfx1250) — compile-verified
//
#include <hip/hip_runtime.h>
#include <stdint.h>

#define T_LEN     32768
#define OT        1024              // outputs per block
#define XSEG      ((OT + 16) / 4)   // 260 b128 segments incl. 8-elem halo each side
#define NTHREADS  256
#define NEG_SLOPE 0.01f

__device__ __forceinline__ float lrelu(float y) {
    return y >= 0.f ? y : NEG_SLOPE * y;
}

__global__ __launch_bounds__(NTHREADS) void flrelu_fused_kernel(
    const float* __restrict__ x,
    const float* __restrict__ upf,
    const float* __restrict__ dnf,
    float* __restrict__ out)
{
    __shared__ __align__(16) float s_x[OT + 16];   // 1040 f32
    __shared__ __align__(16) float s_ze[OT + 8];   // ze[i] = z[2i-8], i in [0, OT+7]
    __shared__ __align__(16) float s_zo[OT + 8];   // zo[i] = z[2i-7], i in [0, OT+6]

    const int t    = threadIdx.x;
    const int n0   = blockIdx.x * OT;                 // tile start within row
    const int row  = blockIdx.y;                      // b*C + c
    const float* xrow = x   + (size_t)row * T_LEN;
    float*       orow = out + (size_t)row * T_LEN;

    // ---- filters -> uniform registers (SGPRs); fold UP gain (x2) into up taps;
    //      split down filter into even/odd phases ----
    float e[9], o[8], de[9], dо;
    (void)dо;
    float dev[9], dov[8];
#pragma unroll
    for (int j = 0; j < 9;  ++j) e[j]   = 2.f * upf[2*j];
#pragma unroll
    for (int j = 0; j < 8;  ++j) o[j]   = 2.f * upf[2*j + 1];
#pragma unroll
    for (int j = 0; j < 9;  ++j) dev[j] = dnf[2*j];
#pragma unroll
    for (int j = 0; j < 8;  ++j) dov[j] = dnf[2*j + 1];
    (void)de;

    // ---- stage x tile (with halo) into LDS: async b128 for in-range segments,
    //      zero-fill fully-OOB halo segments (halo = 8 => segments never partial) ----
    const uint32_t sx_base = (uint32_t)(uintptr_t)(&s_x[0]);
    for (int i = t; i < XSEG; i += NTHREADS) {
        const int xg = n0 - 8 + 4*i;                  // element index within row
        if (xg >= 0 && xg + 3 < T_LEN) {
            const float*   gp = xrow + xg;            // 16B-aligned global address
            const uint32_t lp = sx_base + 16u * (uint32_t)i;
            asm volatile("global_load_async_to_lds_b128 %0, %1, off"
                         :: "v"(lp), "v"(gp) : "memory");
        } else {
            float4 z4 = {0.f, 0.f, 0.f, 0.f};
            *(float4*)(&s_x[4*i]) = z4;
        }
    }
    asm volatile("s_wait_asynccnt 0x0" ::: "memory");
    __syncthreads();

    // ---- polyphase upsample (x2) + leaky_relu -> phase-split z in LDS ----
    {
        const int b4 = 4*t;
        float X[12];
#pragma unroll
        for (int j = 0; j < 12; ++j) X[j] = s_x[b4 + j];

        float zev[4], zov[4];
#pragma unroll
        for (int i = 0; i < 4; ++i) {
            float ae = 0.f, ao = 0.f;
#pragma unroll
            for (int j = 0; j < 9; ++j) ae += e[j] * X[i + j];
#pragma unroll
            for (int j = 0; j < 8; ++j) ao += o[j] * X[i + 1 + j];
            zev[i] = lrelu(ae);
            zov[i] = lrelu(ao);
        }
        float4 a = {zev[0], zev[1], zev[2], zev[3]};
        float4 b = {zov[0], zov[1], zov[2], zov[3]};
        *(float4*)(&s_ze[4*t]) = a;   // 16B lane stride: conflict-free ds_store_b128
        *(float4*)(&s_zo[4*t]) = b;

        // halo: ze[1024..1031] (threads 0..7), zo[1024..1030] (threads 8..14)
        if (t < 8) {
            const int i = OT + t;
            float ae = 0.f;
#pragma unroll
            for (int j = 0; j < 9; ++j) ae += e[j] * s_x[i + j];       // max 1031+8=1039
            s_ze[i] = lrelu(ae);
        } else if (t < 15) {
            const int i = OT + (t - 8);
            float ao = 0.f;
#pragma unroll
            for (int j = 0; j < 8; ++j) ao += o[j] * s_x[i + 1 + j];   // max 1030+1+7=1038
            s_zo[i] = lrelu(ao);
        }
    }
    __syncthreads();

    // ---- 17-tap downsample (stride 2) in phase-split form:
    //      out[nl] = sum_j dn[2j]*ze[nl+j] + sum_j dn[2j+1]*zo[nl+j] ----
    {
        float zeb[12], zob[11];
#pragma unroll
        for (int k = 0; k < 12; ++k) zeb[k] = s_ze[4*t + k];
#pragma unroll
        for (int k = 0; k < 11; ++k) zob[k] = s_zo[4*t + k];

        float a0 = 0.f, a1 = 0.f, a2 = 0.f, a3 = 0.f;
#pragma unroll
        for (int j = 0; j < 9; ++j) {
            const float dj = dev[j];
            a0 += dj * zeb[j];
            a1 += dj * zeb[j + 1];
            a2 += dj * zeb[j + 2];
            a3 += dj * zeb[j + 3];
        }
#pragma unroll
        for (int j = 0; j < 8; ++j) {
            const float dj = dov[j];
            a0 += dj * zob[j];
            a1 += dj * zob[j + 1];
            a2 += dj * zob[j + 2];
            a3 += dj * zob[j + 3];
        }
        float4 r = {a0, a1, a2, a3};
        *(float4*)(&orow[n0 + 4*t]) = r;              // coalesced global_store_b128
    }
}

extern "C" void kernel_launch(void* const* d_in, const int* in_sizes, int n_in,
                              void* d_out, int out_size, void* d_ws, size_t ws_size,
                              hipStream_t stream) {
    (void)n_in; (void)out_size; (void)d_ws; (void)ws_size;
    const float* x   = (const float*)d_in[0];
    const float* upf = (const float*)d_in[1];
    const float* dnf = (const float*)d_in[2];
    float* out       = (float*)d_out;

    const int rows = in_sizes[0] / T_LEN;             // B*C = 2048
    dim3 grid(T_LEN / OT, rows, 1);                   // 32 x 2048 blocks
    dim3 block(NTHREADS, 1, 1);
    hipLaunchKernelGGL(flrelu_fused_kernel, grid, block, 0, stream,
                       x, upf, dnf, out);
}